// CapsuleLayer_64819646431577
// MI455X (gfx1250) — compile-verified
//
#include <hip/hip_runtime.h>
#include <hip/hip_bf16.h>

// CapsuleLayer dynamic routing, fused for MI455X (gfx1250, wave32).
//   x: [64][2048][16] f32, W: [2048][32][32][16] f32, out v: [64][32][32] f32
// W (128MB) + x (8MB) stay L2-resident; u_hat (512MB) is never materialized.
// Each routing pass recomputes u_hat with V_WMMA_F32_16X16X4_F32.
// Routing weights are folded into the WMMA B-fragment (per-lane = per-column
// scale), so the WMMA accumulator itself carries s_partial = sum_r w*u_hat
// across the r-chunk — no LDS accumulation, no atomics, deterministic.

#define B_   64
#define R_   2048
#define C_   32
#define O_   32
#define IN_  16
#define RCHUNK 16
#define NCHUNK (R_ / RCHUNK)   // 128

typedef __attribute__((ext_vector_type(2))) float v2f;
typedef __attribute__((ext_vector_type(4))) float f4v;
typedef __attribute__((ext_vector_type(8))) float v8f;

// One 16x16 tile of u_hat^T for fixed (r, c, o-base), chained on acc:
//   rows M = o (obase..obase+15), cols N = b (16 batch rows), K = IN = 16.
// A-frag (16x4 f32): lane L holds row M=L%16, K-pair (2*(L/16), 2*(L/16)+1)
//   -> two consecutive W elements, one b64 load.
// B-frag (4x16 f32): lane L holds col N=L%16; VGPR0 = K=(L/16), VGPR1 = K=2+(L/16).
__device__ __forceinline__ v8f uhat_acc(const float* __restrict__ W,
                                        int r, int c, int obase,
                                        const v2f* __restrict__ bf, v8f acc) {
  const int lane = threadIdx.x & 31;
  const int lh = lane & 15;
  const int hi = lane >> 4;
  const float* wrow =
      W + (((size_t)r * C_ + c) * O_ + (obase + lh)) * IN_ + hi * 2;
#pragma unroll
  for (int s = 0; s < 4; ++s) {
    v2f a = *(const v2f*)(wrow + 4 * s);
    acc = __builtin_amdgcn_wmma_f32_16x16x4_f32(
        /*neg_a=*/false, a, /*neg_b=*/false, bf[s],
        /*c_mod=*/(short)0, acc, /*reuse_a=*/false, /*reuse_b=*/false);
  }
  return acc;
}

// mode 0: coupling = 1/32 uniform (softmax of zero logits), no b-logit I/O.
// mode 1: d = u_hat.v  -> write b-logits, weights = softmax(d).
// mode 2: d = b_old + u_hat.v, weights = softmax(d), no write (last iter).
__global__ void __launch_bounds__(128)
caps_pass(int mode, const float* __restrict__ x, const float* __restrict__ W,
          const float* __restrict__ vin, float* __restrict__ bbuf,
          float* __restrict__ partial) {
  __shared__ float smem[4 * RCHUNK * C_ * 16];  // per-wave weights [r][c][b16]
  const int wave = threadIdx.x >> 5;
  const int lane = threadIdx.x & 31;
  const int lh = lane & 15;
  const int hi = lane >> 4;
  const int bbase = wave * 16;       // 4 waves cover all 64 batch rows
  const int chunk = blockIdx.x;
  float* wgtl = smem + wave * (RCHUNK * C_ * 16);
  const int brow = bbase + lh;

  // ---- Stage A (modes 1,2): logits -> softmax weights in LDS ----
  if (mode != 0) {
    for (int rr = 0; rr < RCHUNK; ++rr) {
      const int r = chunk * RCHUNK + rr;
      if (rr + 1 < RCHUNK) {  // prefetch next r's 64KB W block
        const char* pf =
            (const char*)(W + (size_t)(r + 1) * C_ * O_ * IN_) + lane * 2048;
        __builtin_prefetch(pf, 0, 1);
      }
      const f4v* xp = (const f4v*)(x + ((size_t)brow * R_ + r) * IN_);
      f4v x0 = xp[0], x1 = xp[1], x2 = xp[2], x3 = xp[3];
      v2f bf[4];
      bf[0].x = hi ? x0.y : x0.x;  bf[0].y = hi ? x0.w : x0.z;
      bf[1].x = hi ? x1.y : x1.x;  bf[1].y = hi ? x1.w : x1.z;
      bf[2].x = hi ? x2.y : x2.x;  bf[2].y = hi ? x2.w : x2.z;
      bf[3].x = hi ? x3.y : x3.x;  bf[3].y = hi ? x3.w : x3.z;

      float mrun = -1e30f, srun = 0.0f;
      for (int c = 0; c < C_; ++c) {
        v8f z = {};
        v8f C0 = uhat_acc(W, r, c, 0, bf, z);
        v8f C1 = uhat_acc(W, r, c, 16, bf, z);
        const float* vb = vin + ((size_t)brow * C_ + c) * O_ + hi * 8;
        f4v va = *(const f4v*)vb;
        f4v vbq = *(const f4v*)(vb + 4);
        f4v vc = *(const f4v*)(vb + 16);
        f4v vd = *(const f4v*)(vb + 20);
        float p = C0[0] * va.x + C0[1] * va.y + C0[2] * va.z + C0[3] * va.w +
                  C0[4] * vbq.x + C0[5] * vbq.y + C0[6] * vbq.z + C0[7] * vbq.w +
                  C1[0] * vc.x + C1[1] * vc.y + C1[2] * vc.z + C1[3] * vc.w +
                  C1[4] * vd.x + C1[5] * vd.y + C1[6] * vd.z + C1[7] * vd.w;
        p += __shfl_xor(p, 16, 32);  // combine o-halves (lanes L and L^16)
        float d = p;
        const size_t bidx = ((size_t)r * B_ + brow) * C_ + c;
        if (mode == 2) d += bbuf[bidx];
        if (mode == 1 && hi == 0) bbuf[bidx] = d;
        if (hi == 0) wgtl[(rr * C_ + c) * 16 + lh] = d;
        float mn = fmaxf(mrun, d);  // online softmax
        srun = srun * __expf(mrun - mn) + __expf(d - mn);
        mrun = mn;
      }
      // convert stored logits to weights; half-lanes split the 32 capsules
      const float winv = __builtin_amdgcn_rcpf(srun);
      for (int cc = 0; cc < 16; ++cc) {
        const int c = cc + hi * 16;
        const int a = (rr * C_ + c) * 16 + lh;
        wgtl[a] = __expf(wgtl[a] - mrun) * winv;
      }
    }
  }

  // ---- Stage B: s_partial accumulates inside the WMMA accumulator ----
  const float oscale = (mode == 0) ? 0.03125f : 1.0f;
  for (int c = 0; c < C_; ++c) {
    v8f acc0 = {}, acc1 = {};
    for (int rr = 0; rr < RCHUNK; ++rr) {
      const int r = chunk * RCHUNK + rr;
      if (rr + 1 < RCHUNK) {  // prefetch next r's 2KB W slice for this c
        const char* pf =
            (const char*)(W + (((size_t)(r + 1) * C_ + c) * O_) * IN_) +
            lane * 64;
        __builtin_prefetch(pf, 0, 1);
      }
      const f4v* xp = (const f4v*)(x + ((size_t)brow * R_ + r) * IN_);
      f4v x0 = xp[0], x1 = xp[1], x2 = xp[2], x3 = xp[3];
      const float wgt =
          (mode == 0) ? 1.0f : wgtl[(rr * C_ + c) * 16 + lh];
      v2f bf[4];  // B-fragment pre-scaled by routing weight (per-column)
      bf[0].x = (hi ? x0.y : x0.x) * wgt;  bf[0].y = (hi ? x0.w : x0.z) * wgt;
      bf[1].x = (hi ? x1.y : x1.x) * wgt;  bf[1].y = (hi ? x1.w : x1.z) * wgt;
      bf[2].x = (hi ? x2.y : x2.x) * wgt;  bf[2].y = (hi ? x2.w : x2.z) * wgt;
      bf[3].x = (hi ? x3.y : x3.x) * wgt;  bf[3].y = (hi ? x3.w : x3.z) * wgt;
      acc0 = uhat_acc(W, r, c, 0, bf, acc0);
      acc1 = uhat_acc(W, r, c, 16, bf, acc1);
    }
    // deterministic flush: partial[chunk][co(1024)][b(64)]
    const size_t base0 = (size_t)chunk * 1024 + c * 32 + hi * 8;
#pragma unroll
    for (int j = 0; j < 8; ++j) {
      partial[(base0 + j) * B_ + brow] = acc0[j] * oscale;
      partial[(base0 + 16 + j) * B_ + brow] = acc1[j] * oscale;
    }
  }
}

// s[co][b] = sum over 128 chunks (fully coalesced reads)
__global__ void caps_reduce(const float* __restrict__ partial,
                            float* __restrict__ sbuf) {
  const int t = blockIdx.x * blockDim.x + threadIdx.x;  // 65536
  float a = 0.0f;
  for (int k = 0; k < NCHUNK; ++k) a += partial[(size_t)k * 65536 + t];
  sbuf[t] = a;
}

// v[b][c][o] = squash(s) ; one thread per (b,c)
__global__ void caps_squash(const float* __restrict__ sbuf,
                            float* __restrict__ vout) {
  const int t = blockIdx.x * blockDim.x + threadIdx.x;  // 2048
  const int b = t >> 5, c = t & 31;
  float sv[32];
  float nn = 0.0f;
#pragma unroll
  for (int o = 0; o < O_; ++o) {
    float e = sbuf[(c * O_ + o) * B_ + b];
    sv[o] = e;
    nn += e * e;
  }
  const float n = sqrtf(nn);
  const float scale = (nn / (1.0f + nn)) / (n + 1e-8f);
#pragma unroll
  for (int o = 0; o < O_; ++o) vout[((size_t)b * C_ + c) * O_ + o] = scale * sv[o];
}

extern "C" void kernel_launch(void* const* d_in, const int* in_sizes, int n_in,
                              void* d_out, int out_size, void* d_ws,
                              size_t ws_size, hipStream_t stream) {
  const float* x = (const float*)d_in[0];
  const float* W = (const float*)d_in[1];
  float* out = (float*)d_out;

  char* ws = (char*)d_ws;
  float* partial = (float*)ws;                                   // 32 MB
  float* bbuf = (float*)(ws + (size_t)NCHUNK * 65536 * 4);       // 16 MB
  float* sbuf = bbuf + (size_t)R_ * B_ * C_;                     // 256 KB
  float* vbuf = sbuf + 65536;                                    // 256 KB

  // iter 0: uniform coupling
  caps_pass<<<NCHUNK, 128, 0, stream>>>(0, x, W, vbuf, bbuf, partial);
  caps_reduce<<<256, 256, 0, stream>>>(partial, sbuf);
  caps_squash<<<8, 256, 0, stream>>>(sbuf, vbuf);
  // iter 1: b = u_hat.v0
  caps_pass<<<NCHUNK, 128, 0, stream>>>(1, x, W, vbuf, bbuf, partial);
  caps_reduce<<<256, 256, 0, stream>>>(partial, sbuf);
  caps_squash<<<8, 256, 0, stream>>>(sbuf, vbuf);
  // iter 2: b += u_hat.v1, final v -> d_out
  caps_pass<<<NCHUNK, 128, 0, stream>>>(2, x, W, vbuf, bbuf, partial);
  caps_reduce<<<256, 256, 0, stream>>>(partial, sbuf);
  caps_squash<<<8, 256, 0, stream>>>(sbuf, out);
}